// ProposalLayer_77395310673972
// MI455X (gfx1250) — compile-verified
//
#include <hip/hip_runtime.h>
#include <stdint.h>

#define NB_BUCKETS 16384
#define PRE_NMS    6000
#define POST_NMS   300
#define CAND_CAP   8192
#define P_PAD      6016          // 376 * 16
#define N_WORDS    376           // 16-bit col-tiles per row, stored as uint32
#define NMS_T      0.7f

typedef float v2f __attribute__((ext_vector_type(2)));
typedef float v8f __attribute__((ext_vector_type(8)));

struct DecBox { float x1, y1, x2, y2; int valid; };

__device__ __forceinline__ DecBox decode_box(int i,
        const float4* __restrict__ anchors,
        const float4* __restrict__ deltas,
        const float*  __restrict__ im_info)
{
    float4 a = anchors[i];
    float4 d = deltas[i];
    float w  = a.z - a.x + 1.0f;
    float h  = a.w - a.y + 1.0f;
    float cx = a.x + 0.5f * w;
    float cy = a.y + 0.5f * h;
    float pcx = d.x * w + cx;
    float pcy = d.y * h + cy;
    float pw  = expf(d.z) * w;
    float ph  = expf(d.w) * h;
    float x1 = pcx - 0.5f * pw;
    float y1 = pcy - 0.5f * ph;
    float x2 = pcx + 0.5f * pw;
    float y2 = pcy + 0.5f * ph;
    float Him = im_info[0], Wim = im_info[1], sc = im_info[2];
    x1 = fminf(fmaxf(x1, 0.0f), Wim - 1.0f);
    x2 = fminf(fmaxf(x2, 0.0f), Wim - 1.0f);
    y1 = fminf(fmaxf(y1, 0.0f), Him - 1.0f);
    y2 = fminf(fmaxf(y2, 0.0f), Him - 1.0f);
    float bw = x2 - x1 + 1.0f;
    float bh = y2 - y1 + 1.0f;
    float ms = 16.0f * sc;
    DecBox r;
    r.x1 = x1; r.y1 = y1; r.x2 = x2; r.y2 = y2;
    r.valid = (bw >= ms) && (bh >= ms);
    return r;
}

// ---------------------------------------------------------------- init
__global__ void k_init(unsigned* __restrict__ hist, int* __restrict__ ctrl,
                       float* __restrict__ out)
{
    int i = blockIdx.x * blockDim.x + threadIdx.x;
    if (i < NB_BUCKETS) hist[i] = 0u;
    if (i < 8)          ctrl[i] = 0;
    if (i < POST_NMS * 4) out[i] = 0.0f;
}

// ------------------------------------------------- pass 1: histogram
__global__ void k_score_hist(const float4* __restrict__ deltas,
                             const float*  __restrict__ probs,
                             const float4* __restrict__ anchors,
                             const float*  __restrict__ im_info,
                             unsigned* __restrict__ hist, int N)
{
    int i = blockIdx.x * blockDim.x + threadIdx.x;
    if (i >= N) return;
    DecBox b = decode_box(i, anchors, deltas, im_info);
    if (!b.valid) return;
    float s = probs[i];                       // valid score in [0,1)
    int bk = (int)(s * (float)NB_BUCKETS);
    bk = min(max(bk, 0), NB_BUCKETS - 1);
    atomicAdd(&hist[bk], 1u);
}

// -------------------------------------------- find threshold bucket b*
__global__ __launch_bounds__(1024)
void k_threshold(const unsigned* __restrict__ hist, int* __restrict__ ctrl)
{
    __shared__ unsigned chunk[1024];
    int t = threadIdx.x;
    unsigned s = 0;
    int b0 = t * 16;
    for (int b = b0; b < b0 + 16; ++b) s += hist[b];
    chunk[t] = s;
    __syncthreads();
    if (t == 0) {
        long long acc = 0;
        int bstar = 0;
        int c;
        for (c = 1023; c >= 0; --c) {
            if (acc + (long long)chunk[c] >= PRE_NMS) break;
            acc += (long long)chunk[c];
        }
        if (c >= 0) {
            int b;
            for (b = c * 16 + 15; b >= c * 16; --b) {
                acc += (long long)hist[b];
                if (acc >= PRE_NMS) break;
            }
            bstar = (b < c * 16) ? c * 16 : b;
        }
        ctrl[0] = bstar;
        ctrl[1] = 0;     // candidate counter (K3 runs after this)
    }
}

// ------------------------------------- pass 2: collect candidate keys
__global__ void k_collect(const float4* __restrict__ deltas,
                          const float*  __restrict__ probs,
                          const float4* __restrict__ anchors,
                          const float*  __restrict__ im_info,
                          int* __restrict__ ctrl,
                          unsigned long long* __restrict__ cand, int N)
{
    int i = blockIdx.x * blockDim.x + threadIdx.x;
    if (i >= N) return;
    DecBox b = decode_box(i, anchors, deltas, im_info);
    if (!b.valid) return;
    float s = probs[i];
    int bk = (int)(s * (float)NB_BUCKETS);
    bk = min(max(bk, 0), NB_BUCKETS - 1);
    if (bk >= ctrl[0]) {
        unsigned pos = atomicAdd((unsigned*)&ctrl[1], 1u);
        if (pos < CAND_CAP) {
            unsigned ord = __float_as_uint(s) | 0x80000000u;   // s >= 0
            cand[pos] = ((unsigned long long)ord << 32)
                      | (unsigned long long)(0xFFFFFFFFu - (unsigned)i);
        }
    }
}

// -------------------- single-block bitonic sort + decode top-6000 boxes
__global__ __launch_bounds__(1024)
void k_sort_emit(const float4* __restrict__ deltas,
                 const float4* __restrict__ anchors,
                 const float*  __restrict__ im_info,
                 const int* __restrict__ ctrl,
                 const unsigned long long* __restrict__ cand,
                 float4* __restrict__ boxes,
                 float*  __restrict__ areas,
                 int*    __restrict__ validf)
{
    __shared__ unsigned long long sk[CAND_CAP];   // 64 KB LDS
    int t = threadIdx.x;
    int Mc = ctrl[1]; if (Mc > CAND_CAP) Mc = CAND_CAP;
    // sort ascending on ~key  ==  descending on key
    for (int s = t; s < CAND_CAP; s += 1024)
        sk[s] = (s < Mc) ? ~cand[s] : ~0ULL;
    __syncthreads();
    for (int k = 2; k <= CAND_CAP; k <<= 1) {
        for (int j = k >> 1; j > 0; j >>= 1) {
            for (int e = t; e < CAND_CAP; e += 1024) {
                int p = e ^ j;
                if (p > e) {
                    unsigned long long x = sk[e], y = sk[p];
                    bool up = ((e & k) == 0);
                    if ((x > y) == up) { sk[e] = y; sk[p] = x; }
                }
            }
            __syncthreads();
        }
    }
    for (int s = t; s < P_PAD; s += 1024) {
        bool ok = (s < PRE_NMS) && (s < Mc);
        if (ok) {
            unsigned long long key = ~sk[s];
            unsigned idx = 0xFFFFFFFFu - (unsigned)(key & 0xFFFFFFFFull);
            DecBox b = decode_box((int)idx, anchors, deltas, im_info);
            boxes[s]  = make_float4(b.x1, b.y1, b.x2, b.y2);
            areas[s]  = (b.x2 - b.x1) * (b.y2 - b.y1);
            validf[s] = 1;
        } else {
            boxes[s]  = make_float4(0.f, 0.f, 0.f, 0.f);
            areas[s]  = 0.f;
            validf[s] = 0;
        }
    }
}

// --------------- suppression bitmask: one wave per 16x16 tile, WMMA for
// --------------- the pairwise area-sum (IoU denominator outer sum)
__global__ __launch_bounds__(32)
void k_iou_mask(const float4* __restrict__ boxes,
                const float*  __restrict__ areas,
                unsigned* __restrict__ mask)
{
    int ct   = blockIdx.x;   // col tile (j)
    int rt   = blockIdx.y;   // row tile (i)
    int lane = threadIdx.x;  // wave32

    if (ct < rt) {           // all j < i here: no suppression bits, but
        if (lane < 16)       // words must be written (ws is poisoned)
            mask[(size_t)(rt * 16 + lane) * N_WORDS + ct] = 0u;
        return;
    }

    __shared__ float4 rows[16];
    float ra = 0.0f;
    if (lane < 16) {
        rows[lane] = boxes[rt * 16 + lane];
        ra         = areas[rt * 16 + lane];
    }
    int c    = ct * 16 + (lane & 15);
    float4 cb = boxes[c];
    float  ca = areas[c];
    __syncthreads();

    // D[M][N] = rowArea[M] + colArea[N] via V_WMMA_F32_16X16X4_F32:
    //   A(16x4): K0 = rowArea, K1 = 1, K2 = K3 = 0
    //   B(4x16): K0 = 1,      K1 = colArea, K2 = K3 = 0
    // A layout: lanes 0-15 hold {K0,K1}, lanes 16-31 hold {K2,K3}
    // B layout: lanes 0-15 hold rows {K0,K1} (col = lane), lanes 16-31 rows {K2,K3}
    bool lo = (lane < 16);
    v2f A, B;
    A.x = lo ? ra   : 0.0f;  A.y = lo ? 1.0f : 0.0f;
    B.x = lo ? 1.0f : 0.0f;  B.y = lo ? ca   : 0.0f;
    v8f C = {};
    v8f D = __builtin_amdgcn_wmma_f32_16x16x4_f32(false, A, false, B,
                                                  (short)0, C, false, false);
    // D layout: lane n<16 -> rows 0..7 (col n); lane n>=16 -> rows 8..15 (col n-16)
    unsigned myMask = 0u;
    #pragma unroll
    for (int r = 0; r < 8; ++r) {
        int rloc = r + (lo ? 0 : 8);
        float4 rb = rows[rloc];
        float ltx = fmaxf(rb.x, cb.x);
        float lty = fmaxf(rb.y, cb.y);
        float rbx = fminf(rb.z, cb.z);
        float rby = fminf(rb.w, cb.w);
        float wx  = fmaxf(rbx - ltx, 0.0f);
        float wy  = fmaxf(rby - lty, 0.0f);
        float inter = wx * wy;
        float S   = D[r];                       // area_i + area_j (exact)
        float iou = inter / (S - inter);        // 0/0 -> NaN -> no suppress
        int gi = rt * 16 + rloc;
        int gj = ct * 16 + (lane & 15);
        bool sup = (iou > NMS_T) && (gj > gi);
        unsigned bal = __builtin_amdgcn_ballot_w32(sup);
        if (lane == r)     myMask = bal & 0xFFFFu;   // row r,   cols 0-15
        if (lane == r + 8) myMask = bal >> 16;       // row r+8, cols 0-15
    }
    if (lane < 16)
        mask[(size_t)(rt * 16 + lane) * N_WORDS + ct] = myMask;
}

// --------------------- greedy scan over the bitmask + rank compaction
__global__ __launch_bounds__(384)
void k_nms_scan(const unsigned* __restrict__ mask,
                const int* __restrict__ validf,
                const float4* __restrict__ boxes,
                float* __restrict__ out)
{
    __shared__ unsigned      remv[N_WORDS];
    __shared__ unsigned char keepf[PRE_NMS];
    __shared__ int           cnts[384];
    int t = threadIdx.x;
    for (int w = t; w < N_WORDS; w += 384) remv[w] = 0u;
    __syncthreads();
    for (int i = 0; i < PRE_NMS; ++i) {
        unsigned wv = remv[i >> 4];
        bool kp = (validf[i] != 0) && (((wv >> (i & 15)) & 1u) == 0u);
        if (t < N_WORDS)   // hide latency of the next row's OR loads
            __builtin_prefetch(&mask[(size_t)(i + 1) * N_WORDS + t], 0, 1);
        __syncthreads();   // all reads of remv done before writes
        if (kp && t < N_WORDS)
            remv[t] |= mask[(size_t)i * N_WORDS + t];
        if (t == 0) keepf[i] = kp ? 1 : 0;
        __syncthreads();   // writes visible before next read
    }
    // parallel rank compaction (chunks of 16 per thread)
    int base = t * 16;
    int cnt = 0;
    for (int s = base; s < base + 16 && s < PRE_NMS; ++s) cnt += keepf[s];
    cnts[t] = cnt;
    __syncthreads();
    if (t == 0) {
        int acc = 0;
        for (int q = 0; q < 384; ++q) { int v = cnts[q]; cnts[q] = acc; acc += v; }
    }
    __syncthreads();
    int rank = cnts[t];
    for (int s = base; s < base + 16 && s < PRE_NMS; ++s) {
        if (keepf[s]) {
            if (rank < POST_NMS) {
                float4 b = boxes[s];
                out[rank * 4 + 0] = b.x;
                out[rank * 4 + 1] = b.y;
                out[rank * 4 + 2] = b.z;
                out[rank * 4 + 3] = b.w;
            }
            rank++;
        }
    }
}

// ---------------------------------------------------------------- host
extern "C" void kernel_launch(void* const* d_in, const int* in_sizes, int n_in,
                              void* d_out, int out_size, void* d_ws, size_t ws_size,
                              hipStream_t stream)
{
    const float4* deltas  = (const float4*)d_in[0];  // rpn_bbox_pred (N,4)
    const float*  probs   = (const float*) d_in[1];  // fg_probs (N)
    const float4* anchors = (const float4*)d_in[2];  // anchors (N,4)
    const float*  iminfo  = (const float*) d_in[3];  // im_info (3)
    float* out = (float*)d_out;
    int N = in_sizes[1];

    char* ws = (char*)d_ws;
    unsigned*           hist   = (unsigned*)          (ws + 0);        // 64 KB
    int*                ctrl   = (int*)               (ws + 65536);    // 256 B
    unsigned long long* cand   = (unsigned long long*)(ws + 65792);    // 64 KB
    float4*             boxes  = (float4*)            (ws + 131328);   // 94 KB
    float*              areas  = (float*)             (ws + 227584);   // 23.5 KB
    int*                validf = (int*)               (ws + 251648);   // 23.5 KB
    unsigned*           mask   = (unsigned*)          (ws + 275712);   // ~8.6 MB

    k_init<<<64, 256, 0, stream>>>(hist, ctrl, out);
    int nb = (N + 255) / 256;
    k_score_hist<<<nb, 256, 0, stream>>>(deltas, probs, anchors, iminfo, hist, N);
    k_threshold<<<1, 1024, 0, stream>>>(hist, ctrl);
    k_collect<<<nb, 256, 0, stream>>>(deltas, probs, anchors, iminfo, ctrl, cand, N);
    k_sort_emit<<<1, 1024, 0, stream>>>(deltas, anchors, iminfo, ctrl, cand,
                                        boxes, areas, validf);
    dim3 gmask(N_WORDS, N_WORDS);
    k_iou_mask<<<gmask, 32, 0, stream>>>(boxes, areas, mask);
    k_nms_scan<<<1, 384, 0, stream>>>(mask, validf, boxes, out);
}